// BatchHardTripletLoss_57698590655314
// MI455X (gfx1250) — compile-verified
//
#include <hip/hip_runtime.h>
#include <hip/hip_bf16.h>

typedef __attribute__((ext_vector_type(2))) float v2f;
typedef __attribute__((ext_vector_type(8))) float v8f;

#define BATCH 4096
#define DIM 128
#define MARGIN_F 0.3f

// ---------------- kernel 0: squared norms -----------------------------------
__global__ void sqnorm_kernel(const float* __restrict__ E, float* __restrict__ sqn) {
    int i = blockIdx.x * blockDim.x + threadIdx.x;
    if (i >= BATCH) return;
    const float4* row = (const float4*)(E + (size_t)i * DIM);
    float s = 0.f;
#pragma unroll
    for (int k = 0; k < DIM / 4; ++k) {
        float4 v = row[k];
        s += v.x * v.x + v.y * v.y + v.z * v.z + v.w * v.w;
    }
    sqn[i] = s;
}

// ---------------- kernel 1: fused WMMA distances + hardest pos/neg ----------
// Block = 256 threads = 8 wave32. Block owns rows [i0, i0+16).
// Wave w scans column tiles j = (w + 8*t)*16, t = 0..31.
// Per 16x16 tile: Gram via 32 chained V_WMMA_F32_16X16X4_F32 (K=128),
// then dist = sqrt(max(sqn_i + sqn_j - 2g, 1e-12)), masked running max/min.
__global__ __launch_bounds__(256) void hardest_kernel(
    const float* __restrict__ E, const long long* __restrict__ labels,
    const float* __restrict__ sqn, float* __restrict__ HP, float* __restrict__ HN)
{
    __shared__ float lds_hp[8 * 16];
    __shared__ float lds_hn[8 * 16];

    const int tid  = threadIdx.x;
    const int wave = tid >> 5;
    const int lane = tid & 31;
    const int half = lane >> 4;     // 0: lanes 0-15, 1: lanes 16-31
    const int l16  = lane & 15;
    const int koff = half * 2;      // K offset within each group of 4 (A and B frags)
    const int roff = half * 8;      // C-frag row offset for this half-wave

    const int i0 = blockIdx.x * 16;

    // Preload A fragments for rows i0..i0+15 (this lane covers row i0+l16).
    v2f afrag[32];
    const float* arow = E + (size_t)(i0 + l16) * DIM;
#pragma unroll
    for (int kk = 0; kk < 32; ++kk)
        afrag[kk] = *(const v2f*)(arow + 4 * kk + koff);

    // Metadata for this lane's 8 C rows: i0 + r + roff.
    long long rowlab[8];
    float rowsq[8];
#pragma unroll
    for (int r = 0; r < 8; ++r) {
        rowlab[r] = labels[i0 + r + roff];
        rowsq[r]  = sqn[i0 + r + roff];
    }

    float hp[8], hn[8];
#pragma unroll
    for (int r = 0; r < 8; ++r) { hp[r] = -1.f; hn[r] = 1e30f; }

    for (int t = 0; t < BATCH / (16 * 8); ++t) {
        const int j   = (wave + 8 * t) * 16;
        const int col = j + l16;                 // this lane's column
        const float* brow      = E + (size_t)col * DIM;   // B = E^T -> load E[col][k]
        const long long collab = labels[col];
        const float colsq      = sqn[col];

        v8f acc = {};
#pragma unroll
        for (int kk = 0; kk < 32; ++kk) {
            v2f bfrag = *(const v2f*)(brow + 4 * kk + koff);
            acc = __builtin_amdgcn_wmma_f32_16x16x4_f32(
                false, afrag[kk], false, bfrag, (short)0, acc, false, false);
        }

#pragma unroll
        for (int r = 0; r < 8; ++r) {
            float sq   = rowsq[r] + colsq - 2.f * acc[r];
            float dist = sqrtf(fmaxf(sq, 1e-12f));
            const int row  = i0 + r + roff;
            const bool same = (rowlab[r] == collab);
            if (same) {
                if (row != col) hp[r] = fmaxf(hp[r], dist);
            } else {
                hn[r] = fminf(hn[r], dist);
            }
        }
    }

    // Reduce across the 16 lanes sharing each row (halves stay disjoint for masks <= 8).
#pragma unroll
    for (int r = 0; r < 8; ++r) {
        float vp = hp[r], vn = hn[r];
#pragma unroll
        for (int m = 8; m >= 1; m >>= 1) {
            vp = fmaxf(vp, __shfl_xor(vp, m, 32));
            vn = fminf(vn, __shfl_xor(vn, m, 32));
        }
        if (l16 == 0) {
            lds_hp[wave * 16 + r + roff] = vp;
            lds_hn[wave * 16 + r + roff] = vn;
        }
    }
    __syncthreads();

    // Combine the 8 waves' partials; one thread per row writes the result.
    if (tid < 16) {
        float vp = -1.f, vn = 1e30f;
#pragma unroll
        for (int w = 0; w < 8; ++w) {
            vp = fmaxf(vp, lds_hp[w * 16 + tid]);
            vn = fminf(vn, lds_hn[w * 16 + tid]);
        }
        HP[i0 + tid] = vp;
        HN[i0 + tid] = vn;
    }
}

// ---------------- kernel 2: scalar loss -------------------------------------
__global__ void finalize_kernel(const float* __restrict__ HP,
                                const float* __restrict__ HN,
                                float* __restrict__ out) {
    __shared__ float ss[256];
    __shared__ float sc[256];
    const int tid = threadIdx.x;
    float s = 0.f, c = 0.f;
    for (int i = tid; i < BATCH; i += 256) {
        float hp = HP[i], hn = HN[i];
        bool valid = (hp >= 0.f) && (hn < 1e29f);   // has_pos && has_neg
        if (valid) {
            s += fmaxf(hp - hn + MARGIN_F, 0.f);
            c += 1.f;
        }
    }
    ss[tid] = s; sc[tid] = c;
    __syncthreads();
    for (int ofs = 128; ofs > 0; ofs >>= 1) {
        if (tid < ofs) { ss[tid] += ss[tid + ofs]; sc[tid] += sc[tid + ofs]; }
        __syncthreads();
    }
    if (tid == 0) out[0] = ss[0] / fmaxf(sc[0], 1.f);
}

// ---------------- launch ----------------------------------------------------
extern "C" void kernel_launch(void* const* d_in, const int* in_sizes, int n_in,
                              void* d_out, int out_size, void* d_ws, size_t ws_size,
                              hipStream_t stream) {
    const float*     E      = (const float*)d_in[0];
    const long long* labels = (const long long*)d_in[1];   // reference labels are int64

    float* sqn = (float*)d_ws;          // 4096 f32
    float* HP  = sqn + BATCH;           // 4096 f32
    float* HN  = HP + BATCH;            // 4096 f32
    float* out = (float*)d_out;

    sqnorm_kernel<<<BATCH / 256, 256, 0, stream>>>(E, sqn);
    hardest_kernel<<<BATCH / 16, 256, 0, stream>>>(E, labels, sqn, HP, HN);
    finalize_kernel<<<1, 256, 0, stream>>>(HP, HN, out);
}